// bifurcation_42666205118777
// MI455X (gfx1250) — compile-verified
//
#include <hip/hip_runtime.h>

typedef __attribute__((ext_vector_type(2))) float v2f;
typedef __attribute__((ext_vector_type(8))) float v8f;

#define TAU_C   0.1f
#define DT_C    0.0005f
#define NNET    4096
#define NB      8192
#define WPW     8                 // waves per workgroup
#define THREADS (WPW * 32)
#define JSLICE  (NNET / WPW)      // 512 j's per wave

// Branchless tanh: 1 - 2/(exp(2x)+1), via hw exp2/rcp (v_exp_f32, v_rcp_f32).
// Saturates correctly: x >> 0 -> exp2 -> +inf -> rcp -> 0 -> 1
//                      x << 0 -> exp2 -> 0    -> rcp -> 1 -> -1
__device__ __forceinline__ float fast_tanh(float x) {
    const float TWO_LOG2E = 2.8853900817779268f;   // 2*log2(e)
    float e = __builtin_amdgcn_exp2f(x * TWO_LOG2E);
    return 1.0f - 2.0f * __builtin_amdgcn_rcpf(e + 1.0f);
}

__global__ __launch_bounds__(THREADS)
void bifurcation_kernel(const float* __restrict__ k_t,
                        const float* __restrict__ phase,
                        const float* __restrict__ monodromy,
                        const float* __restrict__ dK_norms,
                        const float* __restrict__ amp_p,
                        const float* __restrict__ w_p,
                        const float* __restrict__ stim,
                        const float* __restrict__ alpha,
                        const float* __restrict__ I,
                        const float* __restrict__ m,
                        const float* __restrict__ n,
                        float* __restrict__ out)
{
    const int tid   = threadIdx.x;
    const int lane  = tid & 31;
    const int wave  = tid >> 5;
    const int b0    = blockIdx.x * 16;
    const int bl    = lane & 15;          // local batch row (B/D column)
    const int b     = b0 + bl;
    const int khalf = (lane >> 4) << 1;   // K-slot base: 0 (lanes 0-15) or 2 (lanes 16-31)
    const int r     = bl;                 // weight-combo row index in A

    const float amp = amp_p[0];
    const float w   = w_p[0];
    const float s0  = stim[0];
    const float s1  = stim[1];

    const float tau_sw = w * TAU_C;
    const float ph     = phase[b];
    const float vosc   = (amp / sqrtf(1.0f + tau_sw * tau_sw)) *
                         sinf(ph - atan2f(tau_sw, 1.0f));
    const float k0 = k_t[b * 2 + 0];
    const float k1 = k_t[b * 2 + 1];

    const float* m0 = m;
    const float* m1 = m + NNET;
    const float* n0 = n;
    const float* n1 = n + NNET;
    const float* I0 = I;
    const float* I1 = I + NNET;
    const float* I2 = I + 2 * NNET;

    const int jbeg = wave * JSLICE;
    const int jend = jbeg + JSLICE;

    v8f c1 = {};   // rows 0,1  : sum_j tx  * {n0,n1}[j]              (per b column)
    v8f c2 = {};   // rows 0..3 : sum_j dphi* {m0n0,m0n1,m1n0,m1n1}[j]

#pragma unroll 2
    for (int jb = jbeg; jb < jend; jb += 4) {
        const int j = jb + khalf;   // this lane's pair of K slots: j, j+1
        float2 m0v = *(const float2*)(m0 + j);
        float2 m1v = *(const float2*)(m1 + j);
        float2 n0v = *(const float2*)(n0 + j);
        float2 n1v = *(const float2*)(n1 + j);
        float2 I0v = *(const float2*)(I0 + j);
        float2 I1v = *(const float2*)(I1 + j);
        float2 I2v = *(const float2*)(I2 + j);

        v2f a1, a2, btx, bdp;
#pragma unroll
        for (int s = 0; s < 2; ++s) {
            const float m0j = s ? m0v.y : m0v.x;
            const float m1j = s ? m1v.y : m1v.x;
            const float n0j = s ? n0v.y : n0v.x;
            const float n1j = s ? n1v.y : n1v.x;
            const float I0j = s ? I0v.y : I0v.x;
            const float I1j = s ? I1v.y : I1v.x;
            const float I2j = s ? I2v.y : I2v.x;

            const float x    = k0 * m0j + k1 * m1j + vosc * I0j + s0 * I1j + s1 * I2j;
            const float tx   = fast_tanh(x);
            const float dphi = 1.0f - tx * tx;
            btx[s] = tx;
            bdp[s] = dphi;

            // A1 rows: 0 -> n0[j], 1 -> n1[j], rest 0
            const float an  = (r == 0) ? n0j : ((r == 1) ? n1j : 0.0f);
            // A2 rows: 0 -> m0*n0, 1 -> m0*n1, 2 -> m1*n0, 3 -> m1*n1, rest 0
            const float mk  = (r < 2) ? m0j : m1j;
            const float nd  = ((r & 1) == 0) ? n0j : n1j;
            const float amn = (r < 4) ? (mk * nd) : 0.0f;
            a1[s] = an;
            a2[s] = amn;
        }
        c1 = __builtin_amdgcn_wmma_f32_16x16x4_f32(false, a1, false, btx,
                                                   (short)0, c1, false, false);
        c2 = __builtin_amdgcn_wmma_f32_16x16x4_f32(false, a2, false, bdp,
                                                   (short)0, c2, false, false);
    }

    // Per-wave partials -> LDS. D layout: VGPR v, lanes 0-15 = row M=v, col N=lane.
    __shared__ float red[WPW][6][16];
    if (lane < 16) {
        red[wave][0][bl] = c1[0];
        red[wave][1][bl] = c1[1];
        red[wave][2][bl] = c2[0];
        red[wave][3][bl] = c2[1];
        red[wave][4][bl] = c2[2];
        red[wave][5][bl] = c2[3];
    }
    __syncthreads();

    if (tid < 16) {
        const int bb = b0 + tid;
        float sum[6];
#pragma unroll
        for (int q = 0; q < 6; ++q) {
            float s = 0.0f;
#pragma unroll
            for (int wv = 0; wv < WPW; ++wv) s += red[wv][q][tid];
            sum[q] = s;
        }
        const float inv_tau = 1.0f / TAU_C;
        const float invN    = 1.0f / (float)NNET;
        const float phb     = phase[bb];
        const float uosc    = amp * sinf(phb);
        const float kt0     = k_t[bb * 2 + 0];
        const float kt1     = k_t[bb * 2 + 1];

        const float dK0 = inv_tau * (-kt0 + sum[0] * invN +
                                     uosc * alpha[0] + s0 * alpha[1] + s1 * alpha[2]);
        const float dK1 = inv_tau * (-kt1 + sum[1] * invN +
                                     uosc * alpha[3] + s0 * alpha[4] + s1 * alpha[5]);
        const float J00 = inv_tau * (-1.0f + sum[2] * invN);
        const float J01 = inv_tau * (         sum[3] * invN);
        const float J10 = inv_tau * (         sum[4] * invN);
        const float J11 = inv_tau * (-1.0f + sum[5] * invN);

        float* k_out   = out;             // [B,2]
        float* ph_out  = out + NB * 2;    // [B]
        float* mo_out  = out + NB * 3;    // [B,2,2]
        float* nrm_out = out + NB * 7;    // [B]

        k_out[bb * 2 + 0] = kt0 + DT_C * dK0;
        k_out[bb * 2 + 1] = kt1 + DT_C * dK1;
        ph_out[bb]        = phb + DT_C * w;

        const float M00 = monodromy[bb * 4 + 0];
        const float M01 = monodromy[bb * 4 + 1];
        const float M10 = monodromy[bb * 4 + 2];
        const float M11 = monodromy[bb * 4 + 3];
        mo_out[bb * 4 + 0] = M00 + DT_C * (J00 * M00 + J01 * M10);
        mo_out[bb * 4 + 1] = M01 + DT_C * (J00 * M01 + J01 * M11);
        mo_out[bb * 4 + 2] = M10 + DT_C * (J10 * M00 + J11 * M10);
        mo_out[bb * 4 + 3] = M11 + DT_C * (J10 * M01 + J11 * M11);

        nrm_out[bb] = dK_norms[bb] + sqrtf(dK0 * dK0 + dK1 * dK1);
    }
}

extern "C" void kernel_launch(void* const* d_in, const int* in_sizes, int n_in,
                              void* d_out, int out_size, void* d_ws, size_t ws_size,
                              hipStream_t stream) {
    (void)in_sizes; (void)n_in; (void)out_size; (void)d_ws; (void)ws_size;
    const float* k_t       = (const float*)d_in[0];
    const float* phase     = (const float*)d_in[1];
    const float* monodromy = (const float*)d_in[2];
    const float* dK_norms  = (const float*)d_in[3];
    const float* amp       = (const float*)d_in[4];
    const float* w         = (const float*)d_in[5];
    const float* stim      = (const float*)d_in[6];
    const float* alpha     = (const float*)d_in[7];
    const float* I         = (const float*)d_in[8];
    const float* m         = (const float*)d_in[9];
    const float* n         = (const float*)d_in[10];
    float* out             = (float*)d_out;

    dim3 grid(NB / 16);
    dim3 block(THREADS);
    bifurcation_kernel<<<grid, block, 0, stream>>>(
        k_t, phase, monodromy, dK_norms, amp, w, stim, alpha, I, m, n, out);
}